// KnowledgeMemoryv3_4964982194880
// MI455X (gfx1250) — compile-verified
//
#include <hip/hip_runtime.h>
#include <math.h>

// Problem constants (B,Q,M,D,H) = (8,128,1024,256,2)
#define BB 8
#define QQ 128
#define MM 1024
#define DD 256
#define HH 2

typedef __attribute__((ext_vector_type(2))) float v2f;
typedef __attribute__((ext_vector_type(8))) float v8f;

// ---------------------------------------------------------------------------
// f32 WMMA tile GEMM helpers. V_WMMA_F32_16X16X4_F32: D(16x16) += A(16x4)*B(4x16)
// A layout (ISA 7.12.2, 32-bit A 16x4): lanes 0-15 row M=lane, v0=K0,v1=K1;
//                                       lanes 16-31 row M=lane-16, v0=K2,v1=K3.
// B layout (mirror, col per lane):      lanes 0-15 col N=lane, v0=K0,v1=K1;
//                                       lanes 16-31 col N=lane-16, v0=K2,v1=K3.
// C/D layout: VGPR j -> (M = j + 8*(lane>>4), N = lane&15).
// ---------------------------------------------------------------------------

// qh = q * Wq^T + bq       q:(B*Q, D)  Wq:(D,D) row-major (o,i)  qh:(B*Q, D)
__global__ void qh_kernel(const float* __restrict__ q,
                          const float* __restrict__ Wq,
                          const float* __restrict__ bq,
                          float* __restrict__ qh)
{
    const int lane = threadIdx.x & 31;
    const int half = lane >> 4;        // 0 or 1
    const int l15  = lane & 15;
    const int row0 = blockIdx.x * 16;  // over B*Q (=1024)
    const int n0   = blockIdx.y * 16;  // over D (=256)

    const float* arow = q  + (size_t)(row0 + l15) * DD;
    const float* brow = Wq + (size_t)(n0   + l15) * DD;   // B[k][n] = Wq[n][k]

    v8f acc = {};
    #pragma unroll 4
    for (int k = 0; k < DD; k += 4) {
        const int kk = k + half * 2;
        v2f a = { arow[kk], arow[kk + 1] };
        v2f b = { brow[kk], brow[kk + 1] };
        acc = __builtin_amdgcn_wmma_f32_16x16x4_f32(
            false, a, false, b, (short)0, acc, false, false);
    }
    const float bias = bq[n0 + l15];
    #pragma unroll
    for (int j = 0; j < 8; ++j) {
        const int r = row0 + j + half * 8;
        qh[(size_t)r * DD + n0 + l15] = acc[j] + bias;
    }
}

// mh = (kb[:,hop]*sel) * Wm^T    kb:(B,H+1,M,D)  sel:(B,M)  mh:(B*M, D)
__global__ void mh_kernel(const float* __restrict__ kb,
                          const float* __restrict__ sel,
                          const float* __restrict__ Wm,
                          float* __restrict__ mh,
                          int hop)
{
    const int lane = threadIdx.x & 31;
    const int half = lane >> 4;
    const int l15  = lane & 15;
    const int row0 = blockIdx.x * 16;  // over B*M (=8192)
    const int n0   = blockIdx.y * 16;

    const int grow = row0 + l15;       // this lane's A row
    const int b    = grow >> 10;       // /M
    const int m    = grow & (MM - 1);
    const float s  = sel[grow];        // row scale fused into A load

    const float* arow = kb + (((size_t)b * (HH + 1) + hop) * MM + m) * DD;
    const float* brow = Wm + (size_t)(n0 + l15) * DD;

    v8f acc = {};
    #pragma unroll 4
    for (int k = 0; k < DD; k += 4) {
        const int kk = k + half * 2;
        v2f a = { arow[kk] * s, arow[kk + 1] * s };
        v2f b2 = { brow[kk], brow[kk + 1] };
        acc = __builtin_amdgcn_wmma_f32_16x16x4_f32(
            false, a, false, b2, (short)0, acc, false, false);
    }
    #pragma unroll
    for (int j = 0; j < 8; ++j) {
        const int r = row0 + j + half * 8;
        mh[(size_t)r * DD + n0 + l15] = acc[j];
    }
}

// attn[b,q,m] = sum_d v[d]*tanh(qh+mh); mask -> -inf; softmax over m -> w
// grid: (Q/4, B), block 256.  Each thread owns 4 m-rows for 4 q's (16 accs).
__global__ void attn_kernel(const float* __restrict__ qh,   // (B*Q, D)
                            const float* __restrict__ mh,   // (B*M, D)
                            const float* __restrict__ vvec, // (D)
                            const unsigned char* __restrict__ mask, // (B, M)
                            float* __restrict__ w)          // (B*Q, M)
{
    __shared__ float s_qh[4][DD];
    __shared__ float s_v[DD];
    __shared__ float s_red[8];

    const int b   = blockIdx.y;
    const int q0  = blockIdx.x * 4;
    const int tid = threadIdx.x;      // 0..255 == D
    const int wid = tid >> 5;
    const int wln = tid & 31;

    s_v[tid] = vvec[tid];
    #pragma unroll
    for (int j = 0; j < 4; ++j)
        s_qh[j][tid] = qh[(size_t)(b * QQ + q0 + j) * DD + tid];
    __syncthreads();

    float acc[4][4];  // [q][i]
    #pragma unroll
    for (int i = 0; i < 4; ++i) {
        const int m = tid + (i << 8);
        const float4* mrow =
            (const float4*)(mh + ((size_t)(b << 10) + m) * DD);
        float a0 = 0.f, a1 = 0.f, a2 = 0.f, a3 = 0.f;
        for (int d4 = 0; d4 < DD / 4; ++d4) {
            const float4 mv = mrow[d4];
            const int d = d4 * 4;
            #pragma unroll
            for (int e = 0; e < 4; ++e) {
                const float mve = (e == 0) ? mv.x : (e == 1) ? mv.y
                                 : (e == 2) ? mv.z : mv.w;
                const float vd = s_v[d + e];
                a0 = fmaf(vd, tanhf(s_qh[0][d + e] + mve), a0);
                a1 = fmaf(vd, tanhf(s_qh[1][d + e] + mve), a1);
                a2 = fmaf(vd, tanhf(s_qh[2][d + e] + mve), a2);
                a3 = fmaf(vd, tanhf(s_qh[3][d + e] + mve), a3);
            }
        }
        const bool mk = mask[(b << 10) + m] != 0;
        const float NEG = -INFINITY;
        acc[0][i] = mk ? NEG : a0;
        acc[1][i] = mk ? NEG : a1;
        acc[2][i] = mk ? NEG : a2;
        acc[3][i] = mk ? NEG : a3;
    }

    // Softmax over m (=1024) per q
    for (int j = 0; j < 4; ++j) {
        float lm = -INFINITY;
        #pragma unroll
        for (int i = 0; i < 4; ++i) lm = fmaxf(lm, acc[j][i]);
        #pragma unroll
        for (int off = 16; off > 0; off >>= 1)
            lm = fmaxf(lm, __shfl_down(lm, off, 32));
        if (wln == 0) s_red[wid] = lm;
        __syncthreads();
        float bmax = s_red[0];
        #pragma unroll
        for (int i = 1; i < 8; ++i) bmax = fmaxf(bmax, s_red[i]);
        __syncthreads();

        float e[4];
        float ls = 0.f;
        #pragma unroll
        for (int i = 0; i < 4; ++i) {
            e[i] = __expf(acc[j][i] - bmax);  // exp(-inf)=0 handles mask
            ls += e[i];
        }
        #pragma unroll
        for (int off = 16; off > 0; off >>= 1)
            ls += __shfl_down(ls, off, 32);
        if (wln == 0) s_red[wid] = ls;
        __syncthreads();
        float bsum = 0.f;
        #pragma unroll
        for (int i = 0; i < 8; ++i) bsum += s_red[i];
        __syncthreads();

        const float inv = 1.0f / bsum;
        float* wrow = w + (size_t)(b * QQ + q0 + j) * MM;
        #pragma unroll
        for (int i = 0; i < 4; ++i)
            wrow[tid + (i << 8)] = e[i] * inv;
    }
}

// o = w * (kb[:,hop+1]*sel) ; q_out = q_in + o ; optionally emit o.
// grid (B*Q/16, D/16), block 32 (one wave per 16x16 tile). K = M = 1024.
__global__ void out_kernel(const float* __restrict__ w,    // (B*Q, M)
                           const float* __restrict__ kb,   // (B,H+1,M,D)
                           const float* __restrict__ sel,  // (B,M)
                           const float* __restrict__ q_in, // (B*Q, D)
                           float* __restrict__ q_out,      // (B*Q, D)
                           float* __restrict__ o_out,      // may be null
                           int hop)
{
    const int lane = threadIdx.x & 31;
    const int half = lane >> 4;
    const int l15  = lane & 15;
    const int row0 = blockIdx.x * 16;  // over B*Q
    const int n0   = blockIdx.y * 16;  // over D

    const int b = row0 >> 7;           // /Q ; tile never crosses batches
    const float* arow  = w  + (size_t)(row0 + l15) * MM;
    const float* kbase = kb + ((size_t)b * (HH + 1) + hop + 1) * MM * DD;
    const float* sb    = sel + (b << 10);
    const int col = n0 + l15;

    v8f acc = {};
    #pragma unroll 2
    for (int k = 0; k < MM; k += 4) {
        const int kk = k + half * 2;
        v2f a = { arow[kk], arow[kk + 1] };
        v2f bmat = { kbase[(size_t)kk * DD + col] * sb[kk],
                     kbase[(size_t)(kk + 1) * DD + col] * sb[kk + 1] };
        acc = __builtin_amdgcn_wmma_f32_16x16x4_f32(
            false, a, false, bmat, (short)0, acc, false, false);
    }
    #pragma unroll
    for (int j = 0; j < 8; ++j) {
        const int r = row0 + j + half * 8;
        const size_t idx = (size_t)r * DD + col;
        const float o = acc[j];
        if (o_out) o_out[idx] = o;
        q_out[idx] = q_in[idx] + o;
    }
}

extern "C" void kernel_launch(void* const* d_in, const int* in_sizes, int n_in,
                              void* d_out, int out_size, void* d_ws, size_t ws_size,
                              hipStream_t stream) {
    const float* query          = (const float*)d_in[0]; // (B,Q,D)
    const float* kb             = (const float*)d_in[1]; // (B,H+1,M,D)
    const float* sel            = (const float*)d_in[2]; // (B,M)
    const unsigned char* mask   = (const unsigned char*)d_in[3]; // (B,M) bool
    const float* Wq             = (const float*)d_in[4]; // (H,D,D)
    const float* bq             = (const float*)d_in[5]; // (H,D)
    const float* Wm             = (const float*)d_in[6]; // (H,D,D)
    const float* v              = (const float*)d_in[7]; // (H,D)

    float* out_o = (float*)d_out;                         // (B,Q,D)
    float* out_w = out_o + (size_t)BB * QQ * DD;          // (B,Q,M)

    float* q_buf  = (float*)d_ws;                         // B*Q*D
    float* qh_buf = q_buf  + (size_t)BB * QQ * DD;        // B*Q*D
    float* mh_buf = qh_buf + (size_t)BB * QQ * DD;        // B*M*D
    float* w_buf  = mh_buf + (size_t)BB * MM * DD;        // B*Q*M

    // q mutates across hops; never touch the input buffer.
    hipMemcpyAsync(q_buf, query, (size_t)BB * QQ * DD * sizeof(float),
                   hipMemcpyDeviceToDevice, stream);

    for (int h = 0; h < HH; ++h) {
        qh_kernel<<<dim3(BB * QQ / 16, DD / 16), 32, 0, stream>>>(
            q_buf, Wq + (size_t)h * DD * DD, bq + (size_t)h * DD, qh_buf);
        mh_kernel<<<dim3(BB * MM / 16, DD / 16), 32, 0, stream>>>(
            kb, sel, Wm + (size_t)h * DD * DD, mh_buf, h);
        float* wp = (h == HH - 1) ? out_w : w_buf;
        attn_kernel<<<dim3(QQ / 4, BB), 256, 0, stream>>>(
            qh_buf, mh_buf, v + (size_t)h * DD, mask, wp);
        float* op = (h == HH - 1) ? out_o : nullptr;
        out_kernel<<<dim3(BB * QQ / 16, DD / 16), 32, 0, stream>>>(
            wp, kb, sel, q_buf, q_buf, op, h);
    }
}